// GAT_71159018160976
// MI455X (gfx1250) — compile-verified
//
#include <hip/hip_runtime.h>
#include <math.h>

typedef __attribute__((ext_vector_type(2))) float v2f;
typedef __attribute__((ext_vector_type(8))) float v8f;

#define GAT_N 100000
#define GAT_E 1600000
#define NEG_SLOPE 0.2f

// ---- float atomic max via signed-max / unsigned-min trick (init = -inf) ----
__device__ __forceinline__ void atomicMaxFloat(float* addr, float val) {
    if (val >= 0.0f) {
        atomicMax((int*)addr, __float_as_int(val));
    } else {
        atomicMin((unsigned int*)addr, __float_as_uint(val));
    }
}

// ---------------------------------------------------------------------------
// out[n,C] = A[n,K] @ W[K,C], f32 WMMA 16x16x4. One wave per 16x16 tile.
// A layout: lane l holds row (l&15), VGPR i -> K = k0 + 2*(l>>4) + i
// B layout: lane l holds col (l&15), VGPR i -> K = k0 + 2*(l>>4) + i
// D layout: VGPR r -> row r + 8*(l>>4), col (l&15)
// Out-of-range rows/cols are CLAMPED (not zero-filled): a GEMM column/row of D
// depends only on the same column of B / row of A, and clamped rows/cols are
// never stored, so no masking (and no EXEC divergence) is needed in the loop.
// ---------------------------------------------------------------------------
template <int K, int C, int COLTILES>
__global__ void wmma_gemm_f32(const float* __restrict__ A,
                              const float* __restrict__ W,
                              float* __restrict__ out, int n) {
    const int wavesPerBlock = blockDim.x >> 5;
    const int wave = blockIdx.x * wavesPerBlock + ((int)threadIdx.x >> 5);
    const int lane = threadIdx.x & 31;
    const int rowTiles = (n + 15) >> 4;
    if (wave >= rowTiles * COLTILES) return;   // uniform per wave: EXEC stays all-1s

    const int rowTile = wave / COLTILES;
    const int colTile = wave - rowTile * COLTILES;
    const int row0 = rowTile << 4;
    const int col0 = colTile << 4;

    const int mr    = lane & 15;
    const int khalf = (lane >> 4) << 1;        // 0 or 2

    int arow = row0 + mr; if (arow >= n) arow = n - 1;   // clamped, never stored
    int bcol = col0 + mr; if (bcol >= C) bcol = C - 1;   // clamped, never stored

    const float* __restrict__ Ap = A + (size_t)arow * K + khalf;   // 8B aligned
    const float* __restrict__ Wp = W + (size_t)khalf * C + bcol;

    v8f acc = {};
    #pragma unroll 4
    for (int k0 = 0; k0 < K; k0 += 4) {
        const v2f a = __builtin_nontemporal_load((const v2f*)(Ap + k0));
        v2f b;
        b.x = Wp[(size_t)(k0 + 0) * C];
        b.y = Wp[(size_t)(k0 + 1) * C];
        acc = __builtin_amdgcn_wmma_f32_16x16x4_f32(false, a, false, b,
                                                    (short)0, acc, false, false);
    }

    const int orow0 = row0 + ((lane >> 4) << 3);   // +0 or +8
    const int ocol  = col0 + mr;
    if (ocol < C) {
        float* __restrict__ op = out + (size_t)orow0 * C + ocol;
        if (row0 + 16 <= n) {
            // Full row tile (always the case when n % 16 == 0): straight-line
            // stores with constant immediate offsets -> single store clause.
            #pragma unroll
            for (int r = 0; r < 8; ++r) op[r * C] = acc[r];
        } else {
            #pragma unroll
            for (int r = 0; r < 8; ++r) {
                if (orow0 + r < n) op[r * C] = acc[r];
            }
        }
    }
}

// ---- per-node: alpha_s/alpha_d dots, init m=-inf, s=0, zero agg row --------
template <int C>
__global__ void node_prep(const float* __restrict__ h,
                          const float* __restrict__ avs,
                          const float* __restrict__ avd,
                          float* __restrict__ as, float* __restrict__ ad,
                          float* __restrict__ m,  float* __restrict__ ssum,
                          float* __restrict__ agg, int n) {
    const int i = blockIdx.x * blockDim.x + threadIdx.x;
    if (i >= n) return;
    const float* hr = h + (size_t)i * C;
    float s_ = 0.0f, d_ = 0.0f;
    #pragma unroll
    for (int c = 0; c < C; ++c) { const float v = hr[c]; s_ += v * avs[c]; d_ += v * avd[c]; }
    as[i] = s_; ad[i] = d_;
    m[i] = -INFINITY; ssum[i] = 0.0f;
    float* ar = agg + (size_t)i * C;
    #pragma unroll
    for (int c = 0; c < C; ++c) ar[c] = 0.0f;
}

// ---- edge pass 1: alpha_e = leaky_relu(as[src]+ad[dst]) + log2(ew); seg-max
__global__ void edge_alpha_max(const long long* __restrict__ esrc,
                               const long long* __restrict__ edst,
                               const float* __restrict__ ew,
                               const float* __restrict__ as,
                               const float* __restrict__ ad,
                               float* __restrict__ alphaE,
                               float* __restrict__ m, int E, int n) {
    const int e = blockIdx.x * blockDim.x + threadIdx.x;
    if (e >= E + n) return;
    int s, d; float lw;
    if (e < E) { s = (int)esrc[e]; d = (int)edst[e]; lw = log2f(ew[e]); }
    else       { s = d = e - E;   lw = 0.0f; }
    float al = as[s] + ad[d];
    al = (al > 0.0f) ? al : NEG_SLOPE * al;
    al += lw;
    alphaE[e] = al;
    atomicMaxFloat(m + d, al);
}

// ---- edge pass 2: e = exp(alpha - m[dst]); seg-sum -------------------------
__global__ void edge_expsum(const long long* __restrict__ edst,
                            const float* __restrict__ m,
                            float* __restrict__ alphaE,
                            float* __restrict__ ssum, int E, int n) {
    const int e = blockIdx.x * blockDim.x + threadIdx.x;
    if (e >= E + n) return;
    const int d = (e < E) ? (int)edst[e] : (e - E);
    const float v = expf(alphaE[e] - m[d]);
    alphaE[e] = v;
    atomicAdd(ssum + d, v);
}

// ---- edge pass 3: agg[dst] += h[src] * (e / s[dst]) ------------------------
template <int C>
__global__ void edge_scatter(const long long* __restrict__ esrc,
                             const long long* __restrict__ edst,
                             const float* __restrict__ alphaE,
                             const float* __restrict__ ssum,
                             const float* __restrict__ h,
                             float* __restrict__ agg, int E, int n) {
    const int e = blockIdx.x * blockDim.x + threadIdx.x;
    if (e >= E + n) return;
    int s, d;
    if (e < E) { s = (int)esrc[e]; d = (int)edst[e]; }
    else       { s = d = e - E; }
    const float a = alphaE[e] / ssum[d];
    const float4* __restrict__ hr4 = (const float4*)(h + (size_t)s * C);  // 16B aligned (C=16,40)
    float* __restrict__ ar = agg + (size_t)d * C;
    #pragma unroll
    for (int q = 0; q < C / 4; ++q) {
        const float4 hv = hr4[q];
        atomicAdd(&ar[4 * q + 0], hv.x * a);
        atomicAdd(&ar[4 * q + 1], hv.y * a);
        atomicAdd(&ar[4 * q + 2], hv.z * a);
        atomicAdd(&ar[4 * q + 3], hv.w * a);
    }
}

// ---- per-element: out = agg + bias (optional relu) -------------------------
template <int C, bool RELU>
__global__ void node_finish(const float* __restrict__ agg,
                            const float* __restrict__ b,
                            float* __restrict__ out, int total) {
    const int t = blockIdx.x * blockDim.x + threadIdx.x;
    if (t >= total) return;
    float v = agg[t] + b[t % C];
    if (RELU) v = (v > 0.0f) ? v : 0.0f;
    out[t] = v;
}

extern "C" void kernel_launch(void* const* d_in, const int* in_sizes, int n_in,
                              void* d_out, int out_size, void* d_ws, size_t ws_size,
                              hipStream_t stream) {
    (void)in_sizes; (void)n_in; (void)out_size; (void)ws_size;
    const float*     x    = (const float*)d_in[0];
    const long long* ei   = (const long long*)d_in[1];   // int64 [2, E]
    const float*     ew   = (const float*)d_in[2];
    const float*     W1   = (const float*)d_in[3];
    const float*     as1  = (const float*)d_in[4];
    const float*     ad1  = (const float*)d_in[5];
    const float*     b1   = (const float*)d_in[6];
    const float*     W2   = (const float*)d_in[7];
    const float*     as2  = (const float*)d_in[8];
    const float*     ad2  = (const float*)d_in[9];
    const float*     b2   = (const float*)d_in[10];

    const long long* esrc = ei;
    const long long* edst = ei + GAT_E;
    const int n = GAT_N, E = GAT_E, nEtot = E + n;

    // Workspace layout (floats). h1 is reused as the post-relu buffer
    // (node_finish reads only agg/b, so overwrite is safe).
    float* ws     = (float*)d_ws;
    float* h1     = ws;  ws += (size_t)n * 16;     // layer-1 features / post-relu
    float* h2     = ws;  ws += (size_t)n * 40;     // layer-2 features
    float* agg    = ws;  ws += (size_t)n * 40;     // segment-sum accumulator (both layers)
    float* alphas = ws;  ws += n;
    float* alphad = ws;  ws += n;
    float* mmax   = ws;  ws += n;
    float* ssum   = ws;  ws += n;
    float* alphaE = ws;  ws += nEtot;

    const dim3 blk(256);
    const int nodeBlocks = (n + 255) / 256;
    const int edgeBlocks = (nEtot + 255) / 256;
    const int rowTiles   = (n + 15) / 16;

    // ---------------- Layer 1: C=16, K=256 ----------------
    {
        const int waves = rowTiles * 1;
        wmma_gemm_f32<256, 16, 1><<<(waves + 7) / 8, 256, 0, stream>>>(x, W1, h1, n);
    }
    node_prep<16><<<nodeBlocks, blk, 0, stream>>>(h1, as1, ad1, alphas, alphad, mmax, ssum, agg, n);
    edge_alpha_max<<<edgeBlocks, blk, 0, stream>>>(esrc, edst, ew, alphas, alphad, alphaE, mmax, E, n);
    edge_expsum<<<edgeBlocks, blk, 0, stream>>>(edst, mmax, alphaE, ssum, E, n);
    edge_scatter<16><<<edgeBlocks, blk, 0, stream>>>(esrc, edst, alphaE, ssum, h1, agg, E, n);
    node_finish<16, true><<<((n * 16) + 255) / 256, blk, 0, stream>>>(agg, b1, h1, n * 16);

    // ---------------- Layer 2: C=40, K=16 -----------------
    {
        const int waves = rowTiles * 3;            // 40 cols -> 16+16+8 (clamped)
        wmma_gemm_f32<16, 40, 3><<<(waves + 7) / 8, 256, 0, stream>>>(h1, W2, h2, n);
    }
    node_prep<40><<<nodeBlocks, blk, 0, stream>>>(h2, as2, ad2, alphas, alphad, mmax, ssum, agg, n);
    edge_alpha_max<<<edgeBlocks, blk, 0, stream>>>(esrc, edst, ew, alphas, alphad, alphaE, mmax, E, n);
    edge_expsum<<<edgeBlocks, blk, 0, stream>>>(edst, mmax, alphaE, ssum, E, n);
    edge_scatter<40><<<edgeBlocks, blk, 0, stream>>>(esrc, edst, alphaE, ssum, h2, agg, E, n);
    node_finish<40, false><<<((n * 40) + 255) / 256, blk, 0, stream>>>(agg, b2, (float*)d_out, n * 40);
}